// LMMD_loss_81698867904837
// MI455X (gfx1250) — compile-verified
//
#include <hip/hip_runtime.h>
#include <math.h>
#include <stdint.h>

typedef __attribute__((ext_vector_type(2))) float v2f;
typedef __attribute__((ext_vector_type(8))) float v8f;

#define B_    4096
#define D_    256
#define C_    31
#define N_TOT 8192

// workspace layout (float offsets)
#define WS_SVEC   0                     // 4096*32 padded normalized s_label
#define WS_TVEC   (4096 * 32)          // 4096*32 padded normalized t_label
#define WS_SQ     (2 * 4096 * 32)      // 8192 row squared norms of [src;tgt]
#define WS_COLSUM (WS_SQ + 8192)       // 256 column sums of total
#define WS_SSUM   (WS_COLSUM + 256)    // 32
#define WS_TSUM   (WS_SSUM + 32)       // 32
#define WS_INVBW  (WS_TSUM + 32)       // 5 inverse bandwidths

// ---------------- prep kernels ----------------

__global__ void k_label_sums(const float* __restrict__ s_label,
                             const float* __restrict__ t_label,
                             float* __restrict__ ws) {
  int t = threadIdx.x;  // 64 threads
  if (t < C_) {
    float a = 0.f;
    for (int r = 0; r < B_; ++r) a += s_label[r * C_ + t];
    ws[WS_SSUM + t] = a;
  } else if (t >= 32 && t < 32 + C_) {
    int c = t - 32;
    float a = 0.f;
    for (int r = 0; r < B_; ++r) a += t_label[r * C_ + c];
    ws[WS_TSUM + c] = a;
  }
  if (t == C_) ws[WS_SSUM + 31] = 0.f;
  if (t == 32 + C_) ws[WS_TSUM + 31] = 0.f;
}

__global__ void k_colsum(const float* __restrict__ src,
                         const float* __restrict__ tgt,
                         float* __restrict__ ws) {
  int d = threadIdx.x;  // 256 threads, one column each (coalesced across threads)
  float a = 0.f;
  for (int r = 0; r < B_; ++r) a += src[(size_t)r * D_ + d];
  for (int r = 0; r < B_; ++r) a += tgt[(size_t)r * D_ + d];
  ws[WS_COLSUM + d] = a;
}

__global__ void k_rownorm(const float* __restrict__ src,
                          const float* __restrict__ tgt,
                          float* __restrict__ ws) {
  int wave = threadIdx.x >> 5, lane = threadIdx.x & 31;
  int row = blockIdx.x * 8 + wave;  // 1024 blocks * 8 waves = 8192 rows
  const float* X = (row < B_) ? (src + (size_t)row * D_)
                              : (tgt + (size_t)(row - B_) * D_);
  float a = 0.f;
  for (int t = 0; t < 8; ++t) {
    float v = X[lane + t * 32];
    a += v * v;
  }
  for (int off = 16; off; off >>= 1) a += __shfl_xor(a, off, 32);
  if (lane == 0) ws[WS_SQ + row] = a;
}

__global__ void k_norm_labels(const float* __restrict__ s_label,
                              const float* __restrict__ t_label,
                              float* __restrict__ ws) {
  int gid = blockIdx.x * blockDim.x + threadIdx.x;  // 2*4096*32 threads
  int half = B_ * 32;
  int which = gid >= half;
  int idx = which ? gid - half : gid;
  int r = idx >> 5, c = idx & 31;
  float out = 0.f;
  if (c < C_) {
    float sS = ws[WS_SSUM + c], tS = ws[WS_TSUM + c];
    if (sS != 0.f && tS != 0.f) {  // valid column; else vec entry is 0
      const float* L = which ? t_label : s_label;
      float denom = which ? tS : sS;
      out = L[r * C_ + c] / denom;
    }
  }
  ws[(which ? WS_TVEC : WS_SVEC) + idx] = out;
}

__global__ void k_bandwidth(float* __restrict__ ws, float* __restrict__ out) {
  __shared__ float red[256];
  int t = threadIdx.x;
  float s = 0.f;
  for (int i = t; i < N_TOT; i += 256) s += ws[WS_SQ + i];
  float c = ws[WS_COLSUM + t];
  red[t] = s;
  __syncthreads();
  for (int off = 128; off; off >>= 1) {
    if (t < off) red[t] += red[t + off];
    __syncthreads();
  }
  float S = red[0];
  __syncthreads();
  red[t] = c * c;
  __syncthreads();
  for (int off = 128; off; off >>= 1) {
    if (t < off) red[t] += red[t + off];
    __syncthreads();
  }
  if (t == 0) {
    float Q = red[0];
    // sum(l2) = 2*n*sum(sq) - 2*||colsum||^2
    float suml2 = 2.f * (float)N_TOT * S - 2.f * Q;
    float bw = suml2 / (float)((long long)N_TOT * N_TOT - N_TOT);
    bw *= 0.25f;  // / KERNEL_MUL^(KERNEL_NUM//2) = / 4
    float m = 1.f;
    for (int k = 0; k < 5; ++k) {
      ws[WS_INVBW + k] = 1.f / (bw * m);
      m *= 2.f;
    }
    out[0] = 0.f;  // accumulation target
  }
}

// ---------------- fused MMD main kernel ----------------
// grid (64, 64, 3); z=0: SS (+1), z=1: TT (+1), z=2: ST (-2)
// Each block: 64x64 pair tile, double-buffered async global->LDS staging.

// Issue one K-chunk (64 wide) of X rows and Y rows as async LDS loads.
// Flat LDS pointer low 32 bits == LDS byte address (ISA 10.2 aperture rule).
__device__ __forceinline__ void stage_async(const float* __restrict__ X,
                                            const float* __restrict__ Y,
                                            float* dstA, float* dstB,
                                            int I0, int J0, int kc, int tid) {
  for (int it = 0; it < 4; ++it) {
    int e = tid + it * 256;  // 1024 float4 groups: 64 rows x 16
    int r = e >> 4, c4 = e & 15;
    const float* gx = &X[(size_t)(I0 + r) * D_ + kc * 64 + c4 * 4];
    const float* gy = &Y[(size_t)(J0 + r) * D_ + kc * 64 + c4 * 4];
    unsigned la = (unsigned)(uintptr_t)&dstA[r * 68 + c4 * 4];
    unsigned lb = (unsigned)(uintptr_t)&dstB[r * 68 + c4 * 4];
    asm volatile("global_load_async_to_lds_b128 %0, %1, off"
                 :: "v"(la), "v"(gx) : "memory");
    asm volatile("global_load_async_to_lds_b128 %0, %1, off"
                 :: "v"(lb), "v"(gy) : "memory");
  }
}

__global__ __launch_bounds__(256) void k_mmd_main(const float* __restrict__ src,
                                                  const float* __restrict__ tgt,
                                                  const float* __restrict__ ws,
                                                  float* __restrict__ out) {
  __shared__ float lData[2][2][64 * 68];  // [buf][A/B], stride 68: bank-free
  __shared__ float lWA[64 * 36];          // X labels (32 padded cols, stride 36)
  __shared__ float lWB[64 * 36];          // Y labels
  __shared__ float partial[8];

  const int z = blockIdx.z;
  const float* X = (z == 1) ? tgt : src;
  const float* Y = (z == 0) ? src : tgt;
  const float* LX = ws + ((z == 1) ? WS_TVEC : WS_SVEC);
  const float* LY = ws + ((z == 0) ? WS_SVEC : WS_TVEC);
  const int sqXoff = WS_SQ + ((z == 1) ? B_ : 0);
  const int sqYoff = WS_SQ + ((z == 0) ? 0 : B_);
  const float sign = (z == 2) ? -2.f : 1.f;

  const int I0 = blockIdx.x * 64, J0 = blockIdx.y * 64;
  const int tid = threadIdx.x;
  const int wave = tid >> 5, lane = tid & 31;
  const int row = lane & 15;            // M (or N) index within fragment
  const int koff = (lane >> 4) * 2;     // K sub-offset per ISA 16x4 f32 A layout

  // kick off async staging of data chunk 0 immediately
  stage_async(X, Y, lData[0][0], lData[0][1], I0, J0, 0, tid);

  // stage label tiles (2048 elements each) with regular stores meanwhile
  for (int it = 0; it < 8; ++it) {
    int e = tid + it * 256;
    int r = e >> 5, c = e & 31;
    lWA[r * 36 + c] = LX[(size_t)(I0 + r) * 32 + c];
    lWB[r * 36 + c] = LY[(size_t)(J0 + r) * 32 + c];
  }
  __syncthreads();

  // two 16x16 tiles per wave in the 4x4 tile grid (tj1 == tj0 -> B frags shared)
  const int ti0 = wave >> 2, tj0 = wave & 3;
  const int ti1 = (wave + 8) >> 2, tj1 = (wave + 8) & 3;

  // label Gram accumulators (K = 32 padded)
  v8f w0 = {}, w1 = {};
  for (int k = 0; k < 32; k += 4) {
    v2f a0 = *(const v2f*)&lWA[(ti0 * 16 + row) * 36 + k + koff];
    v2f b0 = *(const v2f*)&lWB[(tj0 * 16 + row) * 36 + k + koff];
    w0 = __builtin_amdgcn_wmma_f32_16x16x4_f32(false, a0, false, b0, (short)0,
                                               w0, false, false);
    v2f a1 = *(const v2f*)&lWA[(ti1 * 16 + row) * 36 + k + koff];
    v2f b1 = *(const v2f*)&lWB[(tj1 * 16 + row) * 36 + k + koff];
    w1 = __builtin_amdgcn_wmma_f32_16x16x4_f32(false, a1, false, b1, (short)0,
                                               w1, false, false);
  }

  // data dot accumulators over D=256: 4 chunks of K=64, double buffered.
  // Async loads complete in order per wave: waiting asynccnt<=8 after issuing
  // the next chunk's 8 loads guarantees the current chunk's 8 are in LDS.
  v8f d0 = {}, d1 = {};
  for (int kc = 0; kc < 4; ++kc) {
    const int cur = kc & 1;
    if (kc < 3) {
      // buf 1-cur was last read in iteration kc-1; the trailing barrier of
      // that iteration makes overwriting it here safe.
      stage_async(X, Y, lData[1 - cur][0], lData[1 - cur][1], I0, J0, kc + 1,
                  tid);
      asm volatile("s_wait_asynccnt 8" ::: "memory");
    } else {
      asm volatile("s_wait_asynccnt 0" ::: "memory");
    }
    __syncthreads();  // all waves' current-chunk data visible in LDS

    const float* cA = lData[cur][0];
    const float* cB = lData[cur][1];
    for (int k = 0; k < 64; k += 4) {
      v2f a0 = *(const v2f*)&cA[(ti0 * 16 + row) * 68 + k + koff];
      v2f b0 = *(const v2f*)&cB[(tj0 * 16 + row) * 68 + k + koff];
      d0 = __builtin_amdgcn_wmma_f32_16x16x4_f32(false, a0, false, b0, (short)0,
                                                 d0, false, false);
      v2f a1 = *(const v2f*)&cA[(ti1 * 16 + row) * 68 + k + koff];
      v2f b1 = *(const v2f*)&cB[(tj1 * 16 + row) * 68 + k + koff];
      d1 = __builtin_amdgcn_wmma_f32_16x16x4_f32(false, a1, false, b1, (short)0,
                                                 d1, false, false);
    }
    __syncthreads();  // all reads of buf cur done before it is restaged
  }

  // epilogue: l2 -> 5-kernel gaussian sum -> weighted reduce
  const float ib0 = ws[WS_INVBW + 0], ib1 = ws[WS_INVBW + 1],
              ib2 = ws[WS_INVBW + 2], ib3 = ws[WS_INVBW + 3],
              ib4 = ws[WS_INVBW + 4];
  const int mhi = (lane >> 4) * 8;  // C/D layout: VGPR v -> M = v + 8*(lane>>4)
  const int n0 = lane & 15;         // N = lane & 15
  float accum = 0.f;
  {
    float sqj = ws[sqYoff + J0 + tj0 * 16 + n0];
    int ibase = I0 + ti0 * 16 + mhi;
    for (int v = 0; v < 8; ++v) {
      float l2 = ws[sqXoff + ibase + v] + sqj - 2.f * d0[v];
      float kern = expf(-l2 * ib0) + expf(-l2 * ib1) + expf(-l2 * ib2) +
                   expf(-l2 * ib3) + expf(-l2 * ib4);
      accum += w0[v] * kern;
    }
  }
  {
    float sqj = ws[sqYoff + J0 + tj1 * 16 + n0];
    int ibase = I0 + ti1 * 16 + mhi;
    for (int v = 0; v < 8; ++v) {
      float l2 = ws[sqXoff + ibase + v] + sqj - 2.f * d1[v];
      float kern = expf(-l2 * ib0) + expf(-l2 * ib1) + expf(-l2 * ib2) +
                   expf(-l2 * ib3) + expf(-l2 * ib4);
      accum += w1[v] * kern;
    }
  }
  for (int off = 16; off; off >>= 1) accum += __shfl_xor(accum, off, 32);
  if (lane == 0) partial[wave] = accum;
  __syncthreads();
  if (tid == 0) {
    float s = 0.f;
    for (int i = 0; i < 8; ++i) s += partial[i];
    atomicAdd(out, sign * s);
  }
}

// ---------------- launcher ----------------

extern "C" void kernel_launch(void* const* d_in, const int* in_sizes, int n_in,
                              void* d_out, int out_size, void* d_ws,
                              size_t ws_size, hipStream_t stream) {
  (void)in_sizes; (void)n_in; (void)out_size; (void)ws_size;
  const float* src = (const float*)d_in[0];
  const float* tgt = (const float*)d_in[1];
  const float* s_label = (const float*)d_in[2];
  const float* t_label = (const float*)d_in[3];
  float* ws = (float*)d_ws;
  float* out = (float*)d_out;

  k_label_sums<<<1, 64, 0, stream>>>(s_label, t_label, ws);
  k_colsum<<<1, 256, 0, stream>>>(src, tgt, ws);
  k_rownorm<<<1024, 256, 0, stream>>>(src, tgt, ws);
  k_norm_labels<<<1024, 256, 0, stream>>>(s_label, t_label, ws);
  k_bandwidth<<<1, 256, 0, stream>>>(ws, out);  // also zeroes out[0]
  dim3 grid(64, 64, 3);
  k_mmd_main<<<grid, 256, 0, stream>>>(src, tgt, ws, out);
}